// MaxUnpooling2D_80908593922393
// MI455X (gfx1250) — compile-verified
//
#include <hip/hip_runtime.h>

// MaxUnpooling2D for MI455X (gfx1250, wave32).
// B=8, H=W=128, C=128, 2x2 unpool -> out [8,256,256,128] fp32.
//
// Formulation: gather-per-window. Thread = (window, 4 channels).
// Reads its update float4 + mask int4 once, writes the four 2x2 window
// positions (value in selected slot, zeros elsewhere). Every output element
// written exactly once -> no zero-init pass, no atomics.
// Traffic = 64+64 MiB read + 256 MiB write = 384 MiB (roofline-minimal),
// ~16.5 us at 23.3 TB/s.
//
// CDNA5 paths: async global->LDS double-buffered input stream (ASYNCcnt),
// non-temporal b128 loads/stores (single-use streaming data, bypass L2
// retention), wave32-exact tiling (32 lanes x float4 = one C=128 row).

#define BB   8
#define HH   128
#define WW   128
#define CC   128
#define HOUTC 256
#define WOUTC 256

#define THREADS 256
#define WIN_PER_TILE (THREADS / 32)                 // 8 windows per block-tile
#define NWIN   (BB * HH * WW)                       // 131072 windows
#define NTILES (NWIN / WIN_PER_TILE)                // 16384 tiles
#define GRID_BLOCKS 2048                            // 8 tiles per block

typedef float v4f __attribute__((ext_vector_type(4)));
typedef int   v4i __attribute__((ext_vector_type(4)));

#define AS1 __attribute__((address_space(1)))
#define AS3 __attribute__((address_space(3)))

#if __has_builtin(__builtin_amdgcn_global_load_async_to_lds_b128) && \
    __has_builtin(__builtin_amdgcn_s_wait_asynccnt)
#define USE_ASYNC 1
#else
#define USE_ASYNC 0
#endif

#if USE_ASYNC
// 16-byte async DMA: global -> LDS. Builtin params are v4i pointers in
// address spaces 1 (global) and 3 (LDS); offset/cpol must be immediates.
__device__ __forceinline__ void async_copy16(const void* gsrc, void* ldst) {
    __builtin_amdgcn_global_load_async_to_lds_b128(
        (AS1 v4i*)gsrc, (AS3 v4i*)ldst, 0, 0);
}
#endif

__global__ __launch_bounds__(THREADS) void
max_unpool2d_kernel(const float* __restrict__ upd,
                    const int* __restrict__ mask,
                    float* __restrict__ out)
{
#if USE_ASYNC
    __shared__ v4f s_upd[2][THREADS];
    __shared__ v4i s_msk[2][THREADS];
#endif
    const int tid      = threadIdx.x;
    const int lane_win = tid >> 5;          // 0..7: window within tile
    const int lane_c   = (tid & 31) << 2;   // channel offset (x4 per lane)

    int t = (int)blockIdx.x;
    if (t >= NTILES) return;

#if USE_ASYNC
    // Prologue: kick off async DMA of tile t into buffer 0.
    {
        const int wid  = t * WIN_PER_TILE + lane_win;
        const int gidx = wid * CC + lane_c;
        async_copy16(upd  + gidx, &s_upd[0][tid]);
        async_copy16(mask + gidx, &s_msk[0][tid]);
    }
#endif

    int buf = 0;
    while (t < NTILES) {
        const int tnext = t + (int)gridDim.x;

        v4f v;
        v4i m;
#if USE_ASYNC
        if (tnext < NTILES) {
            // Issue next tile's DMA into the other buffer, then wait only for
            // the current tile (the 2 just-issued loads may stay in flight).
            const int wid  = tnext * WIN_PER_TILE + lane_win;
            const int gidx = wid * CC + lane_c;
            async_copy16(upd  + gidx, &s_upd[buf ^ 1][tid]);
            async_copy16(mask + gidx, &s_msk[buf ^ 1][tid]);
            __builtin_amdgcn_s_wait_asynccnt(2);
        } else {
            __builtin_amdgcn_s_wait_asynccnt(0);
        }
        // Each lane reads back exactly the LDS slot its own async load filled:
        // no cross-lane sharing -> no barrier needed, ASYNCcnt wait suffices.
        v = s_upd[buf][tid];
        m = s_msk[buf][tid];
#else
        {
            const int wid  = t * WIN_PER_TILE + lane_win;
            const int gidx = wid * CC + lane_c;
            v = __builtin_nontemporal_load((const v4f*)(upd + gidx));
            m = __builtin_nontemporal_load((const v4i*)(mask + gidx));
        }
#endif
        // Decode window coordinates from linear window id.
        const int wid = t * WIN_PER_TILE + lane_win;
        const int b =  wid >> 14;          // H*W = 16384
        const int h = (wid >> 7) & 127;
        const int w =  wid & 127;
        // Output base for slot (0,0) of this 2x2 window, at our channel group.
        const int base = ((b * HOUTC + 2 * h) * WOUTC + 2 * w) * CC + lane_c;

        v4f o0 = {0.f, 0.f, 0.f, 0.f};
        v4f o1 = o0, o2 = o0, o3 = o0;
#pragma unroll
        for (int k = 0; k < 4; ++k) {
            // mask = (y*WOUT + x)*C + f ; C=128, WOUT=256 (powers of two).
            const int q   = m[k] >> 7;                 // y*256 + x
            const int sel = ((q >> 7) & 2) | (q & 1);  // dy*2 + dx
            const float val = v[k];
            o0[k] = (sel == 0) ? val : 0.f;
            o1[k] = (sel == 1) ? val : 0.f;
            o2[k] = (sel == 2) ? val : 0.f;
            o3[k] = (sel == 3) ? val : 0.f;
        }
        // Four coalesced 512B/wave streaming stores (single-use -> NT hint).
        __builtin_nontemporal_store(o0, (v4f*)(out + base));
        __builtin_nontemporal_store(o1, (v4f*)(out + base + CC));
        __builtin_nontemporal_store(o2, (v4f*)(out + base + WOUTC * CC));
        __builtin_nontemporal_store(o3, (v4f*)(out + base + WOUTC * CC + CC));

        buf ^= 1;
        t = tnext;
    }
}

extern "C" void kernel_launch(void* const* d_in, const int* in_sizes, int n_in,
                              void* d_out, int out_size, void* d_ws, size_t ws_size,
                              hipStream_t stream) {
    (void)in_sizes; (void)n_in; (void)out_size; (void)d_ws; (void)ws_size;
    const float* upd  = (const float*)d_in[0];
    const int*   mask = (const int*)d_in[1];
    float*       out  = (float*)d_out;

    dim3 grid(GRID_BLOCKS), block(THREADS);
    max_unpool2d_kernel<<<grid, block, 0, stream>>>(upd, mask, out);
}